// AttentionMLPPooling_34772055228497
// MI455X (gfx1250) — compile-verified
//
#include <hip/hip_runtime.h>
#include <hip/hip_bf16.h>
#include <math.h>

// Problem constants (from reference)
#define BB   128
#define NN   64
#define HID  128
#define MLP  128
#define DS   32
#define DV   32
#define DH   64
#define FILLV (-10.0f)
#define ROWS (BB*NN)           // 8192
#define INV_SQRT_MLP 0.088388347648318447f   // 1/sqrt(128)

typedef __attribute__((ext_vector_type(2))) float v2f;
typedef __attribute__((ext_vector_type(8))) float v8f;

// ---------------------------------------------------------------------------
// f32 WMMA GEMM, fully specialized at compile time: C = post(A @ B + bias).
// K is always 128 in this problem -> constexpr, unrolled.
// One wave computes one 16x16 tile with V_WMMA_F32_16X16X4_F32:
//   A frag (16x4):  lanes 0-15 row=lid hold K=k+0,k+1; lanes 16-31 hold K=k+2,k+3
//   B frag (4x16):  lanes 0-15 col=lid hold K=k+0,k+1; lanes 16-31 hold K=k+2,k+3
//   C/D (16x16):    VGPR r -> rows r / r+8 (lane halves), col = lid
// Contiguous-direction loads are vectorized to b64 (8B-aligned: lda/ldb even).
// NANM implements _embed_mask: NaN elements -> 0 in the matmul; any-NaN row
// -> fill after relu, row flags combined across lane halves via wave32 ballot.
// ---------------------------------------------------------------------------
template<bool TA, bool TB, bool NANM, bool RELU, bool HASBIAS>
__global__ __launch_bounds__(256)
void gemm_f32_wmma(const float* __restrict__ A, int lda,
                   const float* __restrict__ Bm, int ldb,
                   float* __restrict__ C, int ldc,
                   const float* __restrict__ bias, float fill,
                   int M, int N)
{
    constexpr int KK = 128;
    int wave   = (blockIdx.x * blockDim.x + threadIdx.x) >> 5;
    int lane   = threadIdx.x & 31;
    int tilesN = N >> 4;
    int tiles  = (M >> 4) * tilesN;
    if (wave >= tiles) return;            // uniform whole-wave exit (EXEC all-1s for WMMA)

    int tm  = (wave / tilesN) << 4;
    int tn  = (wave % tilesN) << 4;
    int hf  = lane >> 4;                  // which K-pair this lane holds
    int lid = lane & 15;
    int arow = tm + lid;
    int bcol = tn + lid;

    const float* Arow = TA ? (A + arow) : (A + (size_t)arow * lda);
    const float* Bcol = TB ? (Bm + (size_t)bcol * ldb) : (Bm + bcol);

    v8f acc = {0.f, 0.f, 0.f, 0.f, 0.f, 0.f, 0.f, 0.f};
    bool anan = false;

    #pragma unroll 8
    for (int k = 0; k < KK; k += 4) {
        int ka = k + 2 * hf;
        v2f a, b;
        if (TA) {
            a.x = Arow[(size_t)ka * lda];
            a.y = Arow[(size_t)(ka + 1) * lda];
        } else {
            a = *(const v2f*)(Arow + ka);          // global_load_b64
        }
        if (NANM) {
            anan |= (a.x != a.x) || (a.y != a.y);
            a.x = (a.x != a.x) ? 0.f : a.x;
            a.y = (a.y != a.y) ? 0.f : a.y;
        }
        if (TB) {
            b = *(const v2f*)(Bcol + ka);          // global_load_b64
        } else {
            b.x = Bcol[(size_t)ka * ldb];
            b.y = Bcol[(size_t)(ka + 1) * ldb];
        }
        acc = __builtin_amdgcn_wmma_f32_16x16x4_f32(
            /*neg_a=*/false, a, /*neg_b=*/false, b,
            /*c_mod=*/(short)0, acc, /*reuse_a=*/false, /*reuse_b=*/false);
    }

    unsigned rowmask = 0;
    if (NANM) {
        unsigned long long bm = __ballot(anan);    // wave32: low 32 bits
        rowmask = (unsigned)bm;
    }

    float bval = HASBIAS ? bias[bcol] : 0.0f;
    #pragma unroll
    for (int r = 0; r < 8; ++r) {
        int m = r + 8 * hf;
        float v = acc[r] + bval;
        if (RELU) v = fmaxf(v, 0.0f);
        if (NANM && (((rowmask >> m) | (rowmask >> (m + 16))) & 1u)) v = fill;
        C[(size_t)(tm + m) * ldc + (tn + lid)] = v;
    }
}

// ---------------------------------------------------------------------------
// Build emb_diag rows: emb[b,i,i] = [sp_diag(32) | hid[b,i](64) | dv_diag(32)]
// ---------------------------------------------------------------------------
__global__ void emb_diag_kernel(const float* __restrict__ obs1,
                                const float* __restrict__ obs2,
                                const float* __restrict__ w_sp,
                                const float* __restrict__ b_sp,
                                const float* __restrict__ w_vel,
                                const float* __restrict__ b_vel,
                                const float* __restrict__ hid,
                                float* __restrict__ emb_diag)
{
    int bi = blockIdx.x * blockDim.x + threadIdx.x;
    if (bi >= ROWS) return;
    int base = bi * 2;

    float o2x = obs2[base], o2y = obs2[base + 1];
    float relx = o2x - o2x, rely = o2y - o2y;              // 0, or NaN if obs2 NaN
    bool  sn = (relx != relx) || (rely != rely);
    float mx = (relx != relx) ? 0.f : relx;
    float my = (rely != rely) ? 0.f : rely;

    float vx = o2x - obs1[base], vy = o2y - obs1[base + 1];
    float rvx = (vx - vx) * 4.f, rvy = (vy - vy) * 4.f;    // 0, or NaN
    bool  dn = (rvx != rvx) || (rvy != rvy);
    float rx = (rvx != rvx) ? 0.f : rvx;
    float ry = (rvy != rvy) ? 0.f : rvy;

    float* row = emb_diag + (size_t)bi * MLP;
    #pragma unroll 8
    for (int c = 0; c < DS; ++c) {
        float e = fmaxf(fmaf(mx, w_sp[c], fmaf(my, w_sp[DS + c], b_sp[c])), 0.f);
        row[c] = sn ? FILLV : e;
    }
    const float* h = hid + (size_t)bi * DH;
    #pragma unroll 8
    for (int c = 0; c < DH; ++c) row[DS + c] = h[c];
    #pragma unroll 8
    for (int c = 0; c < DV; ++c) {
        float e = fmaxf(fmaf(rx, w_vel[c], fmaf(ry, w_vel[DV + c], b_vel[c])), 0.f);
        row[DS + DH + c] = dn ? FILLV : e;
    }
}

// ---------------------------------------------------------------------------
// Attention core per (b,i): scores[j] = emb[b,i,j] . g[b,i] (scaled), softmax,
// pooled t[b,i,:] = sum_j attn_j * emb[b,i,j,:]. sp/dv channels regenerated
// from 2 scalars each; hid read from ws. 64 threads per block.
// ---------------------------------------------------------------------------
__global__ __launch_bounds__(64)
void attn_pool_kernel(const float* __restrict__ obs1,
                      const float* __restrict__ obs2,
                      const float* __restrict__ w_sp,
                      const float* __restrict__ b_sp,
                      const float* __restrict__ w_vel,
                      const float* __restrict__ b_vel,
                      const float* __restrict__ hid,
                      const float* __restrict__ g,
                      float* __restrict__ tout)
{
    __shared__ float gs[MLP];
    __shared__ float wsp[2 * DS], bsp[DS], wvl[2 * DV], bvl[DV];
    __shared__ float mxs[NN], mys[NN], rxs[NN], rys[NN];
    __shared__ int   sns[NN], dns[NN];
    __shared__ float sc[NN];

    int bi = blockIdx.x;
    int b = bi >> 6, i = bi & 63;
    int t = threadIdx.x;                  // 0..63

    gs[t]      = g[(size_t)bi * MLP + t];
    gs[64 + t] = g[(size_t)bi * MLP + 64 + t];
    if (t < 32) {
        wsp[t] = w_sp[t]; wsp[32 + t] = w_sp[32 + t]; bsp[t] = b_sp[t];
        wvl[t] = w_vel[t]; wvl[32 + t] = w_vel[32 + t]; bvl[t] = b_vel[t];
    }

    int ibase = (b * NN + i) * 2;
    float o2ix = obs2[ibase], o2iy = obs2[ibase + 1];
    float vix = o2ix - obs1[ibase], viy = o2iy - obs1[ibase + 1];

    int j = t;
    int jbase = (b * NN + j) * 2;
    float o2jx = obs2[jbase], o2jy = obs2[jbase + 1];
    float relx = o2jx - o2ix, rely = o2jy - o2iy;
    int   sn = (relx != relx) || (rely != rely);
    float mx = (relx != relx) ? 0.f : relx;
    float my = (rely != rely) ? 0.f : rely;
    float vjx = o2jx - obs1[jbase], vjy = o2jy - obs1[jbase + 1];
    float rvx = (vjx - vix) * 4.f, rvy = (vjy - viy) * 4.f;
    int   dn = (rvx != rvx) || (rvy != rvy);
    float rx = (rvx != rvx) ? 0.f : rvx;
    float ry = (rvy != rvy) ? 0.f : rvy;
    mxs[t] = mx; mys[t] = my; rxs[t] = rx; rys[t] = ry; sns[t] = sn; dns[t] = dn;
    __syncthreads();

    // ---- score for column j (q.b_ik term dropped: softmax-invariant) ----
    const float* hj = hid + (size_t)(b * NN + j) * DH;
    float s = 0.f;
    #pragma unroll 8
    for (int c = 0; c < DS; ++c) {
        float e = sn ? FILLV : fmaxf(fmaf(mx, wsp[c], fmaf(my, wsp[32 + c], bsp[c])), 0.f);
        s = fmaf(e, gs[c], s);
    }
    #pragma unroll 8
    for (int c = 0; c < DH; ++c) s = fmaf(hj[c], gs[DS + c], s);
    #pragma unroll 8
    for (int c = 0; c < DV; ++c) {
        float e = dn ? FILLV : fmaxf(fmaf(rx, wvl[c], fmaf(ry, wvl[32 + c], bvl[c])), 0.f);
        s = fmaf(e, gs[DS + DH + c], s);
    }
    sc[t] = s * INV_SQRT_MLP;
    __syncthreads();

    // ---- softmax over 64 columns ----
    float mval = -3.4e38f;
    for (int jj = 0; jj < NN; ++jj) mval = fmaxf(mval, sc[jj]);
    float ex = expf(sc[t] - mval);
    __syncthreads();
    sc[t] = ex;
    __syncthreads();
    float sum = 0.f;
    for (int jj = 0; jj < NN; ++jj) sum += sc[jj];
    float at = ex / sum;
    __syncthreads();
    sc[t] = at;                           // sc now holds attn weights
    __syncthreads();

    // ---- pooled t[b,i,c] = sum_j attn_j * emb[b,i,j,c]; 2 channels/thread ----
    for (int p = 0; p < 2; ++p) {
        int c = t + 64 * p;
        float acc = 0.f;
        if (c < DS) {
            for (int jj = 0; jj < NN; ++jj) {
                float e = sns[jj] ? FILLV
                        : fmaxf(fmaf(mxs[jj], wsp[c], fmaf(mys[jj], wsp[32 + c], bsp[c])), 0.f);
                acc = fmaf(sc[jj], e, acc);
            }
        } else if (c < DS + DH) {
            const float* hcol = hid + (size_t)b * NN * DH + (c - DS);
            for (int jj = 0; jj < NN; ++jj) acc = fmaf(sc[jj], hcol[(size_t)jj * DH], acc);
        } else {
            int d = c - DS - DH;
            for (int jj = 0; jj < NN; ++jj) {
                float e = dns[jj] ? FILLV
                        : fmaxf(fmaf(rxs[jj], wvl[d], fmaf(rys[jj], wvl[32 + d], bvl[d])), 0.f);
                acc = fmaf(sc[jj], e, acc);
            }
        }
        tout[(size_t)bi * MLP + c] = acc;
    }
}

// ---------------------------------------------------------------------------
static inline int gemm_blocks(int M, int N) {
    int tiles = (M / 16) * (N / 16);      // one wave per tile, 8 waves per block
    return (tiles + 7) / 8;
}

extern "C" void kernel_launch(void* const* d_in, const int* in_sizes, int n_in,
                              void* d_out, int out_size, void* d_ws, size_t ws_size,
                              hipStream_t stream) {
    const float* hidden    = (const float*)d_in[0];
    const float* obs1      = (const float*)d_in[1];
    const float* obs2      = (const float*)d_in[2];
    const float* w_sp      = (const float*)d_in[3];
    const float* b_sp      = (const float*)d_in[4];
    const float* w_vel     = (const float*)d_in[5];
    const float* b_vel     = (const float*)d_in[6];
    const float* w_hid     = (const float*)d_in[7];
    const float* b_hid     = (const float*)d_in[8];
    const float* wq        = (const float*)d_in[9];
    const float* wk        = (const float*)d_in[10];
    const float* wv        = (const float*)d_in[11];
    const float* in_proj_w = (const float*)d_in[12];
    const float* in_proj_b = (const float*)d_in[13];
    const float* mha_out_w = (const float*)d_in[14];
    const float* mha_out_b = (const float*)d_in[15];
    const float* out_w     = (const float*)d_in[16];
    const float* out_b     = (const float*)d_in[17];
    float* out = (float*)d_out;

    // workspace layout (~14.2 MB)
    float* WQ   = (float*)d_ws;                 // 128x128
    float* WKT  = WQ  + MLP * MLP;              // 128x128 (== (wk@w_ik)^T)
    float* WV   = WKT + MLP * MLP;              // 128x128
    float* hid  = WV  + MLP * MLP;              // 8192x64
    float* bufA = hid + (size_t)ROWS * DH;      // emb_diag, later pooled t   (8192x128)
    float* bufB = bufA + (size_t)ROWS * MLP;    // q, later ctx               (8192x128)
    float* bufC = bufB + (size_t)ROWS * MLP;    // g, later mha               (8192x128)

    // 1) WQ = wq @ w_iq                 (w_iq = in_proj_w[:, 0:128], ldb=384)
    gemm_f32_wmma<false,false,false,false,false><<<gemm_blocks(128,128), 256, 0, stream>>>(
        wq, 128, in_proj_w, 384, WQ, 128, nullptr, 0.f, 128, 128);
    // 2) WKT = (wk @ w_ik)^T  ->  WKT[d,c] = sum_e w_ik[e,d]*wk[c,e]
    gemm_f32_wmma<true,true,false,false,false><<<gemm_blocks(128,128), 256, 0, stream>>>(
        in_proj_w + 128, 384, wk, 128, WKT, 128, nullptr, 0.f, 128, 128);
    // 3) WV = wv @ w_iv                 (w_iv = in_proj_w[:, 256:384])
    gemm_f32_wmma<false,false,false,false,false><<<gemm_blocks(128,128), 256, 0, stream>>>(
        wv, 128, in_proj_w + 256, 384, WV, 128, nullptr, 0.f, 128, 128);
    // 4) hid = embed_mask(hidden): relu(hidden @ w_hid + b_hid), NaN-row -> 0
    gemm_f32_wmma<false,false,true,true,true><<<gemm_blocks(ROWS,DH), 256, 0, stream>>>(
        hidden, HID, w_hid, DH, hid, DH, b_hid, 0.0f, ROWS, DH);
    // 5) emb_diag rows -> bufA
    emb_diag_kernel<<<ROWS / 256, 256, 0, stream>>>(
        obs1, obs2, w_sp, b_sp, w_vel, b_vel, hid, bufA);
    // 6) q = emb_diag @ WQ + b_iq      (b_iq = in_proj_b[0:128]) -> bufB
    gemm_f32_wmma<false,false,false,false,true><<<gemm_blocks(ROWS,MLP), 256, 0, stream>>>(
        bufA, MLP, WQ, MLP, bufB, MLP, in_proj_b, 0.f, ROWS, MLP);
    // 7) g = q @ WKT                   -> bufC   (scores = emb . g, b_ik dropped)
    gemm_f32_wmma<false,false,false,false,false><<<gemm_blocks(ROWS,MLP), 256, 0, stream>>>(
        bufB, MLP, WKT, MLP, bufC, MLP, nullptr, 0.f, ROWS, MLP);
    // 8) attention: scores/softmax/pool -> t in bufA (emb_diag dead after 6)
    attn_pool_kernel<<<ROWS, NN, 0, stream>>>(
        obs1, obs2, w_sp, b_sp, w_vel, b_vel, hid, bufC, bufA);
    // 9) ctx = t @ WV + b_iv           -> bufB (q dead after 7)
    gemm_f32_wmma<false,false,false,false,true><<<gemm_blocks(ROWS,MLP), 256, 0, stream>>>(
        bufA, MLP, WV, MLP, bufB, MLP, in_proj_b + 256, 0.f, ROWS, MLP);
    // 10) mha = ctx @ mha_out_w + b    -> bufC (g dead after 8)
    gemm_f32_wmma<false,false,false,false,true><<<gemm_blocks(ROWS,MLP), 256, 0, stream>>>(
        bufB, MLP, mha_out_w, MLP, bufC, MLP, mha_out_b, 0.f, ROWS, MLP);
    // 11) out = mha @ out_w + b        -> d_out
    gemm_f32_wmma<false,false,false,false,true><<<gemm_blocks(ROWS,MLP), 256, 0, stream>>>(
        bufC, MLP, out_w, MLP, out, MLP, out_b, 0.f, ROWS, MLP);
}